// SuperQ_86114094284881
// MI455X (gfx1250) — compile-verified
//
#include <hip/hip_runtime.h>
#include <cstdint>
#include <cstddef>

typedef float v2f __attribute__((ext_vector_type(2)));
typedef float v8f __attribute__((ext_vector_type(8)));

#define MAX_OFFSET_F 0.05f
#define EPS_F 1e-6f

// Raw hardware transcendentals: v_exp_f32 is exp2, v_log_f32 is log2.
#if __has_builtin(__builtin_amdgcn_exp2f)
#define EXP2F(x) __builtin_amdgcn_exp2f(x)
#else
#define EXP2F(x) exp2f(x)
#endif
#if __has_builtin(__builtin_amdgcn_logf)
#define LOG2F(x) __builtin_amdgcn_logf(x)
#else
#define LOG2F(x) __log2f(x)
#endif

// fexp(x,p) = sign(x) * |x|^p   (sign(0) = 0, matching jnp.sign)
// |x|^p = exp2(p * log2|x|): log2(0) = -inf -> exp2(-inf) = 0, consistent.
__device__ __forceinline__ float fexp_f(float x, float p) {
    float r = EXP2F(p * LOG2F(fabsf(x)));
    return x > 0.0f ? r : (x < 0.0f ? -r : 0.0f);
}

// clamp_signed: sgn(v) * max(|v|, EPS), with sgn(0) = -1 per reference
__device__ __forceinline__ float clamp_signed_f(float v) {
    float sgn = v > 0.0f ? 1.0f : -1.0f;
    return sgn * fmaxf(fabsf(v), EPS_F);
}

// Branchless tanh on the hardware transcendental path.
__device__ __forceinline__ float fast_tanh(float x) {
#if __has_builtin(__builtin_amdgcn_tanhf)
    return __builtin_amdgcn_tanhf(x);          // v_tanh_f32 (gfx1250)
#else
    // tanh(x) = sign(x) * (1 - 2/(exp2(2|x|*log2e) + 1)); exp2->inf => 1
    float e = EXP2F(fabsf(x) * 2.885390081777927f);   // 2*log2(e)
#if __has_builtin(__builtin_amdgcn_rcpf)
    float r = 1.0f - 2.0f * __builtin_amdgcn_rcpf(e + 1.0f);
#else
    float r = 1.0f - 2.0f / (e + 1.0f);
#endif
    return copysignf(r, x);
#endif
}

__global__ void __launch_bounds__(128) superq_wmma_kernel(
    const float* __restrict__ sqscale,      // (S,3)
    const float* __restrict__ exponents,    // (S,2)
    const float* __restrict__ translation,  // (S,3)
    const float* __restrict__ rotation,     // (S,3,3)
    const float* __restrict__ etas,         // (S,N)
    const float* __restrict__ omegas,       // (S,N)
    const float* __restrict__ offsets,      // (S*N,3)
    const uint8_t* __restrict__ is_prune,   // (S*N,)
    float* __restrict__ out,                // (S*N,3)
    int S, int N, int chunks)
{
    const int blk  = blockIdx.x;
    const int s    = blk / chunks;
    const int base = (blk - s * chunks) * 128;
    const int tid  = threadIdx.x;
    const int l    = tid & 31;   // lane in wave32
    const int m    = l & 15;     // half-wave row/col index

    // ---- wave-uniform per-superquadric params (scalar loads) ----
    const float a1 = sqscale[(size_t)s * 3 + 0];
    const float a2 = sqscale[(size_t)s * 3 + 1];
    const float a3 = sqscale[(size_t)s * 3 + 2];
    const float e1 = exponents[(size_t)s * 2 + 0];
    const float e2 = exponents[(size_t)s * 2 + 1];
    const float* rotv = rotation    + (size_t)s * 9;
    const float* trv  = translation + (size_t)s * 3;

    // ---- A operand: rows e of [R | t], 16x4 f32 layout ----
    // lanes 0-15:  M=m, VGPR0 = K0 = R[e][0], VGPR1 = K1 = R[e][1]
    // lanes 16-31: M=m, VGPR0 = K2 = R[e][2], VGPR1 = K3 = t[e]
    const int im = (m < 3) ? m : 0;           // safe index, zeroed below
    const float r0 = rotv[im * 3 + 0];
    const float r1 = rotv[im * 3 + 1];
    const float r2 = rotv[im * 3 + 2];
    const float tt = trv[im];
    v2f A;
    A.x = (m < 3) ? ((l < 16) ? r0 : r2) : 0.0f;
    A.y = (m < 3) ? ((l < 16) ? r1 : tt) : 0.0f;

    // ---- per-lane point: compute superquadric surface position ----
    const int n      = base + tid;
    const bool valid = (n < N);
    const size_t sn  = (size_t)s * N + (valid ? n : (N - 1)); // clamp in-bounds
    const float eta = __builtin_nontemporal_load(etas + sn);
    const float om  = __builtin_nontemporal_load(omegas + sn);

    const float ce1 = fexp_f(__cosf(eta), e1);
    const float se1 = fexp_f(__sinf(eta), e1);
    const float co2 = fexp_f(__cosf(om),  e2);
    const float so2 = fexp_f(__sinf(om),  e2);

    const float x = clamp_signed_f(a1 * ce1 * co2);
    const float y = clamp_signed_f(a2 * ce1 * so2);
    const float z = clamp_signed_f(a3 * se1);

    // ---- WMMA: D[e][n] = sum_k [R|t][e][k] * [x y z 1]^T[k][n] ----
    v8f c = {};

    // B (4x16 f32) for points 0..15 of this wave:
    // lanes 0-15: K0=x(own), K1=y(own); lanes 16-31: K2=z(point m), K3=1
    const float z_lo = __shfl(z, m);
    v2f B0;
    B0.x = (l < 16) ? x : z_lo;
    B0.y = (l < 16) ? y : 1.0f;
    v8f d0 = __builtin_amdgcn_wmma_f32_16x16x4_f32(
        false, A, false, B0, (short)0, c, false, false);

    // B for points 16..31: lanes 0-15 fetch x,y from upper half; z is own
    const float x_hi = __shfl(x, m + 16);
    const float y_hi = __shfl(y, m + 16);
    v2f B1;
    B1.x = (l < 16) ? x_hi : z;
    B1.y = (l < 16) ? y_hi : 1.0f;
    v8f d1 = __builtin_amdgcn_wmma_f32_16x16x4_f32(
        false, A, false, B1, (short)0, c, false, false);

    // ---- gather result back to one-lane-per-point layout ----
    // D layout: VGPR v, lanes 0-15 hold (M=v, N=lane). Our e = M = 0..2,
    // so lane n has its point's x',y',z' in d[0..2]. Lanes 16-31 (points
    // 16..31, from d1) fetch from d1's lane (l-16).
    const float g0 = __shfl(d1[0], m);
    const float g1 = __shfl(d1[1], m);
    const float g2 = __shfl(d1[2], m);
    const float gx = (l < 16) ? d0[0] : g0;
    const float gy = (l < 16) ? d0[1] : g1;
    const float gz = (l < 16) ? d0[2] : g2;

    // ---- finalize: tanh offsets, prune mask, coalesced NT store ----
    if (valid) {
        const size_t g = (size_t)s * N + n;
        const size_t o = g * 3;
        const float o0 = __builtin_nontemporal_load(offsets + o + 0);
        const float o1 = __builtin_nontemporal_load(offsets + o + 1);
        const float o2 = __builtin_nontemporal_load(offsets + o + 2);
        const bool pr = __builtin_nontemporal_load(is_prune + g) != 0;
        float fx = gx + fast_tanh(o0) * MAX_OFFSET_F;
        float fy = gy + fast_tanh(o1) * MAX_OFFSET_F;
        float fz = gz + fast_tanh(o2) * MAX_OFFSET_F;
        if (pr) { fx = 0.0f; fy = 0.0f; fz = 0.0f; }
        __builtin_nontemporal_store(fx, out + o + 0);
        __builtin_nontemporal_store(fy, out + o + 1);
        __builtin_nontemporal_store(fz, out + o + 2);
    }
}

extern "C" void kernel_launch(void* const* d_in, const int* in_sizes, int n_in,
                              void* d_out, int out_size, void* d_ws, size_t ws_size,
                              hipStream_t stream) {
    const float*   sqscale     = (const float*)d_in[0];
    const float*   exponents   = (const float*)d_in[1];
    const float*   translation = (const float*)d_in[2];
    const float*   rotation    = (const float*)d_in[3];
    const float*   etas        = (const float*)d_in[4];
    const float*   omegas      = (const float*)d_in[5];
    const float*   offsets     = (const float*)d_in[6];
    const uint8_t* is_prune    = (const uint8_t*)d_in[7];
    float*         out         = (float*)d_out;

    const int S = in_sizes[0] / 3;          // sqscale is (S,3)
    const int N = in_sizes[4] / S;          // etas is (S,N) -> N = 400
    const int chunks = (N + 127) / 128;     // 128-point chunks per block

    dim3 grid((unsigned)(S * chunks)), block(128);
    superq_wmma_kernel<<<grid, block, 0, stream>>>(
        sqscale, exponents, translation, rotation,
        etas, omegas, offsets, is_prune, out, S, N, chunks);
}